// GraphNeuralNetwork_89678917140791
// MI455X (gfx1250) — compile-verified
//
#include <hip/hip_runtime.h>
#include <hip/hip_bf16.h>
#include <cstdint>
#include <cstddef>

#define NNODES 100000
#define NEDGES 1600000
#define DIN    128
#define DHID   128
#define DOUT   64
#define BN_EPS 1e-5f

typedef __attribute__((ext_vector_type(2))) float v2f;
typedef __attribute__((ext_vector_type(8))) float v8f;

__device__ __forceinline__ void atomic_add_f32(float* p, float v) {
    // Hardware global_atomic_add_f32 (no CAS loop).
    unsafeAtomicAdd(p, v);
}

// ---------------- degree / normalization ----------------

__global__ void k_deg_init(float* __restrict__ deg, int n) {
    int i = blockIdx.x * blockDim.x + threadIdx.x;
    if (i < n) deg[i] = 1.0f;            // +1 self-loop
}

__global__ void k_deg_count(const int* __restrict__ dst, float* __restrict__ deg, int ne) {
    int e = blockIdx.x * blockDim.x + threadIdx.x;
    if (e < ne) atomic_add_f32(&deg[dst[e]], 1.0f);
}

__global__ void k_deg_rsqrt(float* __restrict__ deg, int n) {
    int i = blockIdx.x * blockDim.x + threadIdx.x;
    if (i < n) deg[i] = rsqrtf(deg[i]);  // dinv = deg^{-1/2}
}

// ---------------- WMMA f32 GEMM: H[M x ND] = A[M x KD] * W[KD x ND] ----------------
// Block = 256 thr = 8 waves; block covers 128 rows; wave w covers rows row0..row0+15.
// W tile staged in LDS (KD*ND*4 bytes <= 64KB).

template <int KD, int ND>
__global__ __launch_bounds__(256) void k_gemm_wmma_f32(const float* __restrict__ A,
                                                       const float* __restrict__ W,
                                                       float* __restrict__ H, int M) {
    __shared__ float sW[KD * ND];
    for (int i = threadIdx.x; i < KD * ND; i += 256) sW[i] = W[i];
    __syncthreads();

    const int wave = threadIdx.x >> 5;
    const int lane = threadIdx.x & 31;
    const int half = lane >> 4;       // 0: lanes 0-15, 1: lanes 16-31
    const int l16  = lane & 15;
    const int row0 = blockIdx.x * 128 + wave * 16;
    if (row0 >= M) return;            // M % 16 == 0, so valid waves have full tiles

    v8f acc[ND / 16] = {};

    // A frag: VGPR0 = K(2*half), VGPR1 = K(2*half+1), row = row0 + l16
    const float* arow = A + (size_t)(row0 + l16) * KD + 2 * half;

    for (int k0 = 0; k0 < KD; k0 += 4) {
        v2f a;
        a.x = arow[k0];
        a.y = arow[k0 + 1];
        // B frag: VGPR0 = row K(k0+2*half), VGPR1 = row K(k0+2*half+1), col = l16
        const float* wrow = &sW[(k0 + 2 * half) * ND + l16];
#pragma unroll
        for (int ct = 0; ct < ND / 16; ++ct) {
            v2f b;
            b.x = wrow[ct * 16];
            b.y = wrow[ct * 16 + ND];
            acc[ct] = __builtin_amdgcn_wmma_f32_16x16x4_f32(
                false, a, false, b, (short)0, acc[ct], false, false);
        }
    }

    // D layout: VGPR v holds row (row0 + v + 8*half), col l16
    const int orow = row0 + 8 * half;
#pragma unroll
    for (int ct = 0; ct < ND / 16; ++ct) {
#pragma unroll
        for (int v = 0; v < 8; ++v) {
            H[(size_t)(orow + v) * ND + ct * 16 + l16] = acc[ct][v];
        }
    }
}

// ---------------- self-loop message + bias: agg = dinv^2 * H + b ----------------

template <int ND>
__global__ void k_selfloop_bias(const float* __restrict__ H, const float* __restrict__ dinv,
                                const float* __restrict__ b, float* __restrict__ agg, int n) {
    size_t idx = (size_t)blockIdx.x * blockDim.x + threadIdx.x;
    if (idx >= (size_t)n * ND) return;
    int i = (int)(idx / ND);
    int c = (int)(idx % ND);
    float di = dinv[i];
    agg[idx] = di * di * H[idx] + b[c];
}

// ---------------- edge scatter: agg[dst] += dinv[src]*dinv[dst] * H[src] ----------------
// ND/4 threads cooperate per edge (float4 gather + 4 hardware f32 atomics).

template <int ND>
__global__ void k_scatter(const int* __restrict__ src, const int* __restrict__ dst,
                          const float* __restrict__ dinv, const float* __restrict__ H,
                          float* __restrict__ agg, int ne) {
    constexpr int L = ND / 4;
    size_t gid = (size_t)blockIdx.x * blockDim.x + threadIdx.x;
    size_t e   = gid / L;
    int   part = (int)(gid % L);
    if (e >= (size_t)ne) return;
    int   s    = src[e];
    int   d    = dst[e];
    float norm = dinv[s] * dinv[d];
    const float4 hv = *reinterpret_cast<const float4*>(H + (size_t)s * ND + part * 4);
    float* out = agg + (size_t)d * ND + part * 4;
    atomic_add_f32(out + 0, norm * hv.x);
    atomic_add_f32(out + 1, norm * hv.y);
    atomic_add_f32(out + 2, norm * hv.z);
    atomic_add_f32(out + 3, norm * hv.w);
}

// ---------------- fused BatchNorm(eval) + ReLU, in place ----------------

template <int ND>
__global__ void k_bn_relu(float* __restrict__ x, const float* __restrict__ g,
                          const float* __restrict__ be, const float* __restrict__ rm,
                          const float* __restrict__ rv, int n) {
    size_t idx = (size_t)blockIdx.x * blockDim.x + threadIdx.x;
    if (idx >= (size_t)n * ND) return;
    int c = (int)(idx % ND);
    float v = (x[idx] - rm[c]) * (g[c] * rsqrtf(rv[c] + BN_EPS)) + be[c];
    x[idx] = fmaxf(v, 0.0f);
}

// ---------------- host launch ----------------

static inline unsigned blocks_for(long long total, int bs) {
    return (unsigned)((total + bs - 1) / bs);
}

extern "C" void kernel_launch(void* const* d_in, const int* in_sizes, int n_in,
                              void* d_out, int out_size, void* d_ws, size_t ws_size,
                              hipStream_t stream) {
    (void)in_sizes; (void)n_in; (void)out_size; (void)ws_size;

    const float* x   = (const float*)d_in[0];
    const int*   ei  = (const int*)d_in[1];
    const float* W1  = (const float*)d_in[2];
    const float* b1  = (const float*)d_in[3];
    const float* g1  = (const float*)d_in[4];
    const float* be1 = (const float*)d_in[5];
    const float* rm1 = (const float*)d_in[6];
    const float* rv1 = (const float*)d_in[7];
    const float* W2  = (const float*)d_in[8];
    const float* b2  = (const float*)d_in[9];
    const float* g2  = (const float*)d_in[10];
    const float* be2 = (const float*)d_in[11];
    const float* rm2 = (const float*)d_in[12];
    const float* rv2 = (const float*)d_in[13];
    const float* W3  = (const float*)d_in[14];
    const float* b3  = (const float*)d_in[15];
    float*       out = (float*)d_out;

    const int* srcI = ei;            // edge_index[0]
    const int* dstI = ei + NEDGES;   // edge_index[1]

    char* ws = (char*)d_ws;
    float* dinv = (float*)ws;                                    // 400 KB
    float* bufA = (float*)(ws + (1u << 20));                     // GEMM output H (51.2 MB)
    float* bufB = (float*)(ws + (1u << 20) + (size_t)NNODES * DHID * sizeof(float)); // agg

    const int TB = 256;
    const unsigned gNodes  = blocks_for(NNODES, TB);
    const unsigned gEdges  = blocks_for(NEDGES, TB);
    const unsigned gElemH  = blocks_for((long long)NNODES * DHID, TB);
    const unsigned gElemO  = blocks_for((long long)NNODES * DOUT, TB);
    const unsigned gScatH  = blocks_for((long long)NEDGES * (DHID / 4), TB);
    const unsigned gScatO  = blocks_for((long long)NEDGES * (DOUT / 4), TB);
    const unsigned gGemm   = blocks_for(NNODES, 128);

    // degrees (with self-loop) -> dinv = deg^{-1/2}
    k_deg_init<<<gNodes, TB, 0, stream>>>(dinv, NNODES);
    k_deg_count<<<gEdges, TB, 0, stream>>>(dstI, dinv, NEDGES);
    k_deg_rsqrt<<<gNodes, TB, 0, stream>>>(dinv, NNODES);

    // ---- Layer 1 ----
    k_gemm_wmma_f32<DIN, DHID><<<gGemm, TB, 0, stream>>>(x, W1, bufA, NNODES);
    k_selfloop_bias<DHID><<<gElemH, TB, 0, stream>>>(bufA, dinv, b1, bufB, NNODES);
    k_scatter<DHID><<<gScatH, TB, 0, stream>>>(srcI, dstI, dinv, bufA, bufB, NEDGES);
    k_bn_relu<DHID><<<gElemH, TB, 0, stream>>>(bufB, g1, be1, rm1, rv1, NNODES);

    // ---- Layer 2 ----
    k_gemm_wmma_f32<DHID, DHID><<<gGemm, TB, 0, stream>>>(bufB, W2, bufA, NNODES);
    k_selfloop_bias<DHID><<<gElemH, TB, 0, stream>>>(bufA, dinv, b2, bufB, NNODES);
    k_scatter<DHID><<<gScatH, TB, 0, stream>>>(srcI, dstI, dinv, bufA, bufB, NEDGES);
    k_bn_relu<DHID><<<gElemH, TB, 0, stream>>>(bufB, g2, be2, rm2, rv2, NNODES);

    // ---- Layer 3 (output, no BN/ReLU) ----
    k_gemm_wmma_f32<DHID, DOUT><<<gGemm, TB, 0, stream>>>(bufB, W3, bufA, NNODES);
    k_selfloop_bias<DOUT><<<gElemO, TB, 0, stream>>>(bufA, dinv, b3, out, NNODES);
    k_scatter<DOUT><<<gScatO, TB, 0, stream>>>(srcI, dstI, dinv, bufA, out, NEDGES);
}